// MCL_LOG_44590350467563
// MI455X (gfx1250) — compile-verified
//
#include <hip/hip_runtime.h>
#include <math.h>

#define NUM_CLASSES 1000
#define BATCH       32768
#define KLAB        10
#define EPSF        1e-7f
#define ROWS_PER_BLOCK 8
#define NBLOCKS (BATCH / ROWS_PER_BLOCK)   /* 4096 */

typedef __attribute__((ext_vector_type(2))) float v2f;
typedef __attribute__((ext_vector_type(8))) float v8f;

// ---------------- Stage 1: one wave32 per row -------------------------------
// Two passes over the row (max, then sum-exp); the second pass hits WGP$/L2.
// Per-block partial sums written to d_ws (4096 floats), fixed-order -> deterministic.
__global__ __launch_bounds__(256) void cll_stage1(
    const float* __restrict__ outputs,
    const int*   __restrict__ labels,
    float*       __restrict__ partials)
{
    const int lane = threadIdx.x & 31;
    const int wave = threadIdx.x >> 5;
    const int row  = blockIdx.x * ROWS_PER_BLOCK + wave;

    const float*  rowPtr = outputs + (size_t)row * NUM_CLASSES;
    const float4* rowV   = (const float4*)rowPtr;          // 1000 floats = 250 float4, 16B aligned

    // ---- pass 1: row max (coalesced b128 loads + prefetch) ----
    float m = -INFINITY;
    for (int v = lane; v < NUM_CLASSES / 4; v += 32) {
        if (v + 32 < NUM_CLASSES / 4)
            __builtin_prefetch(&rowV[v + 32], 0, 0);       // global_prefetch_b8
        float4 x = rowV[v];
        m = fmaxf(m, fmaxf(fmaxf(x.x, x.y), fmaxf(x.z, x.w)));
    }
    #pragma unroll
    for (int off = 16; off; off >>= 1)
        m = fmaxf(m, __shfl_xor(m, off, 32));

    // ---- pass 2: sum exp(x - m), data now cache-resident ----
    const float L2E  = 1.4426950408889634f;
    const float mref = m * L2E;
    float s0 = 0.f, s1 = 0.f, s2 = 0.f, s3 = 0.f;
    for (int v = lane; v < NUM_CLASSES / 4; v += 32) {
        float4 x = rowV[v];
        s0 += exp2f(fmaf(x.x, L2E, -mref));
        s1 += exp2f(fmaf(x.y, L2E, -mref));
        s2 += exp2f(fmaf(x.z, L2E, -mref));
        s3 += exp2f(fmaf(x.w, L2E, -mref));
    }
    float s = (s0 + s1) + (s2 + s3);
    #pragma unroll
    for (int off = 16; off; off >>= 1)
        s += __shfl_xor(s, off, 32);

    // ---- complementary labels: count (with duplicates), dedup for the mask ----
    int l = -1;
    if (lane < KLAB) l = labels[row * KLAB + lane];
    const bool valid    = (lane < KLAB) && (l >= 0);
    const int  num_comp = __popcll(__ballot(valid));
    bool uniq = valid;
    #pragma unroll
    for (int j = 0; j < KLAB; ++j) {
        int lj = __shfl(l, j, 32);
        if (j < lane && lj >= 0 && lj == l) uniq = false;
    }
    float e = 0.f;
    if (uniq) e = exp2f(fmaf(rowPtr[l], L2E, -mref));
    #pragma unroll
    for (int off = 16; off; off >>= 1)
        e += __shfl_xor(e, off, 32);

    // ---- per-row loss, deterministic block combine ----
    __shared__ float acc[ROWS_PER_BLOCK];
    if (lane == 0) {
        float sum_non = (s - e) / s;                       // = sum of non-comp softmax probs
        float loss    = -__logf(sum_non + EPSF);
        float scale   = (float)(NUM_CLASSES - 1) / (float)(NUM_CLASSES - num_comp);
        acc[wave] = scale * loss;
    }
    __syncthreads();
    if (threadIdx.x == 0) {
        float t = 0.f;
        #pragma unroll
        for (int i = 0; i < ROWS_PER_BLOCK; ++i) t += acc[i];
        partials[blockIdx.x] = t;
    }
}

// ---------------- Stage 2: WMMA reduction of 4096 partials ------------------
// One wave32. A = 16x4 f32 tile of partials, B = all-ones 4x16 =>
// D[i][j] += sum_k A[i][k]. 64 elements per v_wmma_f32_16x16x4_f32, 64 iters.
// Final scalar = sum of D column 0 (lane0 M=0..7 + lane16 M=8..15), times 1/B.
__global__ __launch_bounds__(32) void cll_stage2_wmma(
    const float* __restrict__ partials,
    float*       __restrict__ out)
{
    const int lane = threadIdx.x;
    const int r    = lane & 15;    // A-matrix row M
    const int half = lane >> 4;    // 0 -> K=0,1 ; 1 -> K=2,3

    v8f acc = {};
    v2f ones; ones[0] = 1.f; ones[1] = 1.f;   // all-ones B, layout-independent

    for (int i = 0; i < NBLOCKS / 64; ++i) {  // 64 iterations
        const float* base = partials + i * 64;
        v2f a = *(const v2f*)(base + r * 4 + half * 2);   // A[r][k], k = 2*half..2*half+1
        acc = __builtin_amdgcn_wmma_f32_16x16x4_f32(
            /*neg_a=*/false, a, /*neg_b=*/false, ones,
            /*c_mod=*/(short)0, acc, /*reuse_a=*/false, /*reuse_b=*/false);
    }

    float t = 0.f;
    #pragma unroll
    for (int i = 0; i < 8; ++i) t += acc[i];              // column 0, M = 0..7 (or 8..15 on lane16)
    float hi = __shfl(t, 16, 32);
    if (lane == 0) out[0] = (t + hi) * (1.0f / (float)BATCH);
}

extern "C" void kernel_launch(void* const* d_in, const int* in_sizes, int n_in,
                              void* d_out, int out_size, void* d_ws, size_t ws_size,
                              hipStream_t stream) {
    const float* outputs = (const float*)d_in[0];   // [B, C] float32
    const int*   labels  = (const int*)d_in[1];     // [B, K] int32 (JAX x64 disabled)
    float* partials = (float*)d_ws;                 // 4096 floats of scratch

    cll_stage1<<<NBLOCKS, 256, 0, stream>>>(outputs, labels, partials);
    cll_stage2_wmma<<<1, 32, 0, stream>>>(partials, (float*)d_out);
}